// Tform_79757542687005
// MI455X (gfx1250) — compile-verified
//
#include <hip/hip_runtime.h>
#include <stdint.h>

// Memory-bound per-pixel polynomial color transform (AI ~2.9 FLOP/B).
// Strategy: saturate HBM with a CDNA5 async-to-LDS triple-buffered prefetch
// pipeline (global_load_async_to_lds_b128 + s_wait_asynccnt), b128 LDS reads,
// Horner-factored 27-FMA/pixel compute, and non-temporal b128 stores.

typedef float v4f __attribute__((ext_vector_type(4)));

static constexpr int kThreads  = 256;              // 8 wave32 per block
static constexpr int kWaves    = kThreads / 32;
static constexpr int kIter     = 8;                // pipeline steps per block
static constexpr int kPixStep  = kThreads * 4;     // 1024 pixels per step
static constexpr int kPixBlock = kPixStep * kIter; // 8192 pixels per block

__device__ __forceinline__ float poly10(const float* cf, float r, float g, float b) {
  // out = cf0*r^2 + cf1*rg + cf2*rb + cf3*r + cf4*g^2 + cf5*gb + cf6*g
  //     + cf7*b^2 + cf8*b + cf9   (Horner-factored: 9 FMA)
  float t1 = __builtin_fmaf(cf[2], b, cf[3]);
  t1 = __builtin_fmaf(cf[1], g, t1);
  t1 = __builtin_fmaf(cf[0], r, t1);
  float t2 = __builtin_fmaf(cf[5], b, cf[6]);
  t2 = __builtin_fmaf(cf[4], g, t2);
  float t3 = __builtin_fmaf(cf[7], b, cf[8]);
  float acc = __builtin_fmaf(b, t3, cf[9]);
  acc = __builtin_fmaf(g, t2, acc);
  acc = __builtin_fmaf(r, t1, acc);
  return __builtin_fminf(1.0f, __builtin_fmaxf(-1.0f, acc));
}

__global__ __launch_bounds__(kThreads) void tform_kernel(
    const float* __restrict__ I, const float* __restrict__ W,
    float* __restrict__ O, int HW) {
  // [buffer 0..2][wave][plane r/g/b][lane] : 36 KB LDS, per-wave ownership,
  // no cross-wave sharing => no workgroup barriers needed.
  __shared__ v4f lds[3][kWaves][3][32];

  const int tid  = threadIdx.x;
  const int lane = tid & 31;
  const int wave = tid >> 5;
  const int img  = blockIdx.y;

  const size_t base = (size_t)img * 3u * (size_t)HW;
  const float* Ib = I + base;
  float*       Ob = O + base;
  const int blockPix = blockIdx.x * kPixBlock;

  static constexpr float kBias[3][10] = {
    {-0.0463f,-0.0476f,-0.0464f, 0.0504f,-0.0350f,-0.0475f, 0.0181f,-0.0463f, 0.0503f, 0.0396f},
    {-0.0710f,-0.0736f,-0.0711f, 0.0828f,-0.0632f,-0.0736f, 0.0531f,-0.0710f, 0.0827f,-0.0060f},
    {-0.0890f,-0.0949f,-0.0890f, 0.1096f,-0.0899f,-0.0949f, 0.0872f,-0.0890f, 0.1095f,-0.0520f}};

  // Per-image coefficients: Wt[o][c] = W[img][c*3+o] + bias[o][c].
  // img is uniform per block -> scalar loads.
  const float* Wb = W + (size_t)img * 30u;
  float cf[3][10];
#pragma unroll
  for (int o = 0; o < 3; ++o)
#pragma unroll
    for (int c = 0; c < 10; ++c)
      cf[o][c] = Wb[c * 3 + o] + kBias[o][c];

  const bool full = (blockPix + kPixBlock) <= HW;

  if (full) {
    // Issue one pipeline step: 3 planes x 16B/lane async global->LDS.
    auto issue = [&](int step, int buf) {
      const int p = blockPix + step * kPixStep + tid * 4;
#pragma unroll
      for (int c = 0; c < 3; ++c) {
        const float* gp = Ib + (size_t)c * (size_t)HW + (size_t)p;
        const uint32_t la = (uint32_t)(uintptr_t)(&lds[buf][wave][c][lane]);
        asm volatile("global_load_async_to_lds_b128 %0, %1, off"
                     :: "v"(la), "v"((uint64_t)(uintptr_t)gp)
                     : "memory");
      }
    };

    issue(0, 0);   // prefetch depth 2 (6 async loads in flight)
    issue(1, 1);

#pragma unroll
    for (int i = 0; i < kIter; ++i) {
      // Async loads complete in order: asynccnt<=3 guarantees step i landed.
      if (i < kIter - 1)
        asm volatile("s_wait_asynccnt 0x3" ::: "memory");
      else
        asm volatile("s_wait_asynccnt 0x0" ::: "memory");

      const int buf = i % 3;
      const int p = blockPix + i * kPixStep + tid * 4;

      v4f r = lds[buf][wave][0][lane];   // ds_load_b128 x3
      v4f g = lds[buf][wave][1][lane];
      v4f b = lds[buf][wave][2][lane];

      v4f o0, o1, o2;
#pragma unroll
      for (int k = 0; k < 4; ++k) {
        o0[k] = poly10(cf[0], r[k], g[k], b[k]);
        o1[k] = poly10(cf[1], r[k], g[k], b[k]);
        o2[k] = poly10(cf[2], r[k], g[k], b[k]);
      }
      // Streaming output, never re-read: NT b128 stores keep L2 for the input.
      __builtin_nontemporal_store(o0, (v4f*)(Ob + (size_t)p));
      __builtin_nontemporal_store(o1, (v4f*)(Ob + (size_t)HW + (size_t)p));
      __builtin_nontemporal_store(o2, (v4f*)(Ob + (size_t)2 * (size_t)HW + (size_t)p));

      if (i + 2 < kIter) {
        // Buffer distance 3 + dscnt drain: safe to overwrite buf (i+2)%3.
        asm volatile("s_wait_dscnt 0x0" ::: "memory");
        issue(i + 2, (i + 2) % 3);
      }
    }
  } else {
    // Tail fallback (not taken for the 1024x1024 harness shapes).
    for (int i = 0; i < kIter; ++i) {
      const int p0 = blockPix + i * kPixStep + tid * 4;
      for (int k = 0; k < 4; ++k) {
        const int p = p0 + k;
        if (p < HW) {
          const float r = Ib[p];
          const float g = Ib[(size_t)HW + p];
          const float b = Ib[(size_t)2 * HW + p];
          Ob[p]                    = poly10(cf[0], r, g, b);
          Ob[(size_t)HW + p]       = poly10(cf[1], r, g, b);
          Ob[(size_t)2 * HW + p]   = poly10(cf[2], r, g, b);
        }
      }
    }
  }
}

extern "C" void kernel_launch(void* const* d_in, const int* in_sizes, int n_in,
                              void* d_out, int out_size, void* d_ws, size_t ws_size,
                              hipStream_t stream) {
  const float* I = (const float*)d_in[0];   // (B, 3, H, W) f32
  const float* W = (const float*)d_in[1];   // (B, 30) f32
  float*       O = (float*)d_out;           // (B, 3, H, W) f32

  const int batches = in_sizes[1] / 30;                       // 8
  const long long total = (long long)in_sizes[0];             // B*3*H*W
  const int HW = (int)(total / (3LL * (long long)batches));   // 1048576

  const int gx = (HW + kPixBlock - 1) / kPixBlock;            // 128
  dim3 grid((unsigned)gx, (unsigned)batches);
  dim3 block(kThreads);
  hipLaunchKernelGGL(tform_kernel, grid, block, 0, stream, I, W, O, HW);
}